// Decoder_3925600108976
// MI455X (gfx1250) — compile-verified
//
#include <hip/hip_runtime.h>
#include <cstddef>

// Problem constants (from reference)
#define BB   256
#define MM   128
#define KK   4
#define EE   512
#define VV   32000
#define G3E  1536   // 3*E

typedef __attribute__((ext_vector_type(2))) float v2f;
typedef __attribute__((ext_vector_type(8))) float v8f;

// -------------------------------------------------------------------------
// Gather: out[b*E + e] = table[idx[b]*E + e]   (padding row 0 already zero)
// -------------------------------------------------------------------------
__global__ void gather_rows_kernel(const float* __restrict__ table,
                                   const int* __restrict__ idx,
                                   float* __restrict__ out) {
    int i = blockIdx.x * blockDim.x + threadIdx.x;   // over B*E
    int b = i >> 9;            // /512
    int e = i & (EE - 1);
    out[i] = table[(size_t)idx[b] * EE + e];
}

// -------------------------------------------------------------------------
// Tiled f32 WMMA GEMM:  out[M_rows x N] = A[M_rows x KDIM] @ W[N x KDIM]^T + bias
// Block = 256 threads (8 waves). Block tile: 64 rows x 128 cols.
// Per k-step each wave loads the B fragment ONCE (8B global vector load) and
// 4 A fragments (single ds_load_b64 each, batched ahead of the WMMAs so the
// compiler can clause the loads and issue 4 back-to-back v_wmma ops), then
// runs 4 WMMAs into 4 stacked accumulators.
// LDS row stride = KC+4 floats: 8B-aligned fragments AND conflict-free banks
// (260 mod 64 = 4 -> 16 rows span banks 0,4,...,60; both k-halves disjoint).
// Emits v_wmma_f32_16x16x4_f32 (exact f32 accumulate, matches reference).
// -------------------------------------------------------------------------
template <int KDIM>
__global__ void wmma_gemm_xwt(const float* __restrict__ A,
                              const float* __restrict__ W,
                              const float* __restrict__ bias,
                              float* __restrict__ out,
                              int N) {
    constexpr int KC   = 256;      // K chunk staged in LDS
    constexpr int LDA  = KC + 4;   // padded row stride (floats)
    constexpr int MSUB = 4;        // stacked 16-row M subtiles per wave
    constexpr int MT   = 16 * MSUB;
    __shared__ float ldsA[MT * LDA];   // 64 x 260 f32 = ~66.5 KB

    const int tid  = threadIdx.x;
    const int lane = tid & 31;
    const int wave = tid >> 5;

    const int m0 = blockIdx.x * MT;
    const int n0 = blockIdx.y * 128 + wave * 16;

    // WMMA f32 16x16x4 per-lane fragment coordinates (ISA layout):
    //   A: m = lane&15, k-pair base kk = (lane>>4)*2  -> a = {A[m][kk], A[m][kk+1]}
    //   B: n = lane&15, same kk                       -> b = {B[kk][n], B[kk+1][n]} = W[n][kk..]
    const int mA = lane & 15;
    const int kk = (lane >> 4) * 2;
    const int nc = n0 + (lane & 15);

    const float* __restrict__ wrow_base = W + (size_t)nc * KDIM + kk;

    v8f acc[MSUB];
    #pragma unroll
    for (int ms = 0; ms < MSUB; ++ms) acc[ms] = (v8f){};

    for (int kb = 0; kb < KDIM; kb += KC) {
        // cooperative stage of A tile chunk: MT x 256 floats, coalesced along K
        for (int idx = tid; idx < MT * KC; idx += 256) {
            int r = idx >> 8;          // /256
            int c = idx & (KC - 1);
            ldsA[r * LDA + c] = A[(size_t)(m0 + r) * KDIM + kb + c];
        }
        __syncthreads();

        const float* __restrict__ wrow = wrow_base + kb;
        // prefetch next chunk of the streamed weight row (global_prefetch_b8)
        if (kb + KC < KDIM) __builtin_prefetch(wrow + KC, 0, 1);

        #pragma unroll 2
        for (int k = 0; k < KC; k += 4) {
            // one 8B B-fragment load, shared by 4 WMMAs
            v2f b = *(const v2f*)&wrow[k];
            // batch the 4 A-fragment LDS loads ahead of the WMMAs
            v2f a[MSUB];
            #pragma unroll
            for (int ms = 0; ms < MSUB; ++ms) {
                a[ms] = *(const v2f*)&ldsA[(ms * 16 + mA) * LDA + k + kk];
            }
            #pragma unroll
            for (int ms = 0; ms < MSUB; ++ms) {
                acc[ms] = __builtin_amdgcn_wmma_f32_16x16x4_f32(
                    false, a[ms], false, b, (short)0, acc[ms], false, false);
            }
        }
        __syncthreads();
    }

    // D layout: VGPR r -> row (r + (lane>=16 ? 8 : 0)), col = lane&15
    const int mhi = (lane >> 4) * 8;
    const float bv = bias ? bias[nc] : 0.0f;
    #pragma unroll
    for (int ms = 0; ms < MSUB; ++ms) {
        #pragma unroll
        for (int r = 0; r < 8; ++r) {
            out[(size_t)(m0 + ms * 16 + mhi + r) * N + nc] = acc[ms][r] + bv;
        }
    }
}

// -------------------------------------------------------------------------
// GRU gate combine: h = (1-z)*n + z*h_prev, writes h_out and q(=h)
// -------------------------------------------------------------------------
__global__ void gru_combine_kernel(const float* __restrict__ gi,
                                   const float* __restrict__ gh,
                                   const float* __restrict__ hprev,
                                   float* __restrict__ hout,
                                   float* __restrict__ q) {
    int i = blockIdx.x * blockDim.x + threadIdx.x;   // over B*E
    int b = i >> 9;
    int e = i & (EE - 1);
    const float* gib = gi + (size_t)b * G3E;
    const float* ghb = gh + (size_t)b * G3E;

    float ir = gib[e],           hr = ghb[e];
    float iz = gib[EE + e],      hz = ghb[EE + e];
    float in_ = gib[2 * EE + e], hn = ghb[2 * EE + e];

    float r = 1.0f / (1.0f + __expf(-(ir + hr)));
    float z = 1.0f / (1.0f + __expf(-(iz + hz)));
    float n = tanhf(in_ + r * hn);
    float h = (1.0f - z) * n + z * hprev[i];
    hout[i] = h;
    q[i]    = h;
}

// -------------------------------------------------------------------------
// One memory-network hop, fully fused per batch row (grid = B, block = 256):
//   p[b,m]  = sum_k dot(Tsc[ctx[b,m,k]], q[b])        (gather-dot, no memories)
//   attn    = softmax(p)
//   o[b,e]  = sum_m attn[m] * sum_k To[ctx[b,m,k]][e] (weighted gather-sum)
//   q[b]   += o[b]
// Optional: write raw p (last hop -> p_ptr), write X=[q|o] (hop 0 -> vocab GEMM)
// -------------------------------------------------------------------------
__global__ void hop_kernel(const int* __restrict__ ctx,
                           const float* __restrict__ Tsc,
                           const float* __restrict__ To,
                           float* __restrict__ q,
                           float* __restrict__ p_out,   // nullable
                           float* __restrict__ X) {     // nullable
    __shared__ float qs[EE];
    __shared__ float ps[MM];
    __shared__ float attn[MM];

    const int b    = blockIdx.x;
    const int tid  = threadIdx.x;
    const int lane = tid & 31;
    const int wave = tid >> 5;

    for (int e = tid; e < EE; e += 256) qs[e] = q[(size_t)b * EE + e];
    __syncthreads();

    const int* __restrict__ cb = ctx + (size_t)b * MM * KK;

    // ---- scores: one wave per memory slot (strided), coalesced 128B loads ----
    for (int m = wave; m < MM; m += 8) {
        float acc = 0.0f;
        #pragma unroll
        for (int k = 0; k < KK; ++k) {
            int row = cb[m * KK + k];                 // uniform -> scalar load
            const float* __restrict__ tr = Tsc + (size_t)row * EE;
            for (int e = lane; e < EE; e += 32) acc += tr[e] * qs[e];
        }
        #pragma unroll
        for (int off = 16; off > 0; off >>= 1) acc += __shfl_down(acc, off, 32);
        if (lane == 0) ps[m] = acc;
    }
    __syncthreads();

    if (p_out) {
        for (int m = tid; m < MM; m += 256) p_out[(size_t)b * MM + m] = ps[m];
    }

    // ---- softmax over M=128 (wave 0) ----
    if (wave == 0) {
        float mx = -3.402823466e+38f;
        for (int m = lane; m < MM; m += 32) mx = fmaxf(mx, ps[m]);
        #pragma unroll
        for (int off = 16; off > 0; off >>= 1)
            mx = fmaxf(mx, __shfl_down(mx, off, 32));
        mx = __shfl(mx, 0, 32);
        float sum = 0.0f;
        for (int m = lane; m < MM; m += 32) {
            float ev = __expf(ps[m] - mx);
            attn[m] = ev;
            sum += ev;
        }
        #pragma unroll
        for (int off = 16; off > 0; off >>= 1) sum += __shfl_down(sum, off, 32);
        sum = __shfl(sum, 0, 32);
        float inv = 1.0f / sum;
        for (int m = lane; m < MM; m += 32) attn[m] *= inv;
    }
    __syncthreads();

    // ---- weighted gather-sum for o: thread owns e = tid and tid+256 ----
    float o0 = 0.0f, o1 = 0.0f;
    for (int m = 0; m < MM; ++m) {
        float w = attn[m];
        #pragma unroll
        for (int k = 0; k < KK; ++k) {
            int row = cb[m * KK + k];                 // uniform -> scalar load
            const float* __restrict__ tr = To + (size_t)row * EE;
            o0 += w * tr[tid];
            o1 += w * tr[tid + 256];
        }
    }

    const int e0 = tid, e1 = tid + 256;
    if (X) {   // hop 0: X = [q | o] for vocab projection
        X[(size_t)b * (2 * EE) + e0]      = qs[e0];
        X[(size_t)b * (2 * EE) + e1]      = qs[e1];
        X[(size_t)b * (2 * EE) + EE + e0] = o0;
        X[(size_t)b * (2 * EE) + EE + e1] = o1;
    }
    q[(size_t)b * EE + e0] = qs[e0] + o0;
    q[(size_t)b * EE + e1] = qs[e1] + o1;
}

// -------------------------------------------------------------------------
extern "C" void kernel_launch(void* const* d_in, const int* in_sizes, int n_in,
                              void* d_out, int out_size, void* d_ws, size_t ws_size,
                              hipStream_t stream) {
    const int*   ctx   = (const int*)d_in[0];     // [B,M,K]
    const int*   y_    = (const int*)d_in[1];     // [B]
    const float* h_in  = (const float*)d_in[2];   // [1,B,E]
    const float* A0    = (const float*)d_in[3];   // [V,E]
    const float* A1    = (const float*)d_in[4];
    const float* A2    = (const float*)d_in[5];
    const float* C2    = (const float*)d_in[6];
    const float* W_ih  = (const float*)d_in[7];   // [3E,E]
    const float* W_hh  = (const float*)d_in[8];
    const float* b_ih  = (const float*)d_in[9];   // [3E]
    const float* b_hh  = (const float*)d_in[10];
    const float* lin_W = (const float*)d_in[11];  // [V,2E]
    const float* lin_b = (const float*)d_in[12];  // [V]

    float* out     = (float*)d_out;
    float* p_ptr   = out;                                     // [B*M]
    float* p_vocab = out + (size_t)BB * MM;                   // [B*V]
    float* h_out   = out + (size_t)BB * MM + (size_t)BB * VV; // [B*E]

    float* ws    = (float*)d_ws;
    float* m_emb = ws;                         // B*E
    float* gi    = m_emb + (size_t)BB * EE;    // B*3E
    float* gh    = gi + (size_t)BB * G3E;      // B*3E
    float* qbuf  = gh + (size_t)BB * G3E;      // B*E
    float* Xcat  = qbuf + (size_t)BB * EE;     // B*2E

    // m = A1[y_]  (C[0] tied to A[1])
    gather_rows_kernel<<<(BB * EE) / 256, 256, 0, stream>>>(A1, y_, m_emb);

    // GRU gates via f32 WMMA GEMMs: gi = m @ W_ih^T + b_ih ; gh = h @ W_hh^T + b_hh
    wmma_gemm_xwt<EE><<<dim3(BB / 64, G3E / 128), 256, 0, stream>>>(
        m_emb, W_ih, b_ih, gi, G3E);
    wmma_gemm_xwt<EE><<<dim3(BB / 64, G3E / 128), 256, 0, stream>>>(
        h_in, W_hh, b_hh, gh, G3E);
    gru_combine_kernel<<<(BB * EE) / 256, 256, 0, stream>>>(gi, gh, h_in, h_out, qbuf);

    // 3 memory hops (fused gather/score/softmax/context)
    hop_kernel<<<BB, 256, 0, stream>>>(ctx, A0, A1, qbuf, nullptr, Xcat);
    hop_kernel<<<BB, 256, 0, stream>>>(ctx, A1, A2, qbuf, nullptr, nullptr);
    hop_kernel<<<BB, 256, 0, stream>>>(ctx, A2, C2, qbuf, p_ptr, nullptr);

    // p_vocab = [q|o0] @ lin_W^T + lin_b   (131 MB weight stream, f32 WMMA)
    wmma_gemm_xwt<2 * EE><<<dim3(BB / 64, VV / 128), 256, 0, stream>>>(
        Xcat, lin_W, lin_b, p_vocab, VV);
}